// KentLoss_30210799960108
// MI455X (gfx1250) — compile-verified
//
#include <hip/hip_runtime.h>
#include <math.h>

typedef __attribute__((ext_vector_type(2))) float v2f;
typedef __attribute__((ext_vector_type(8))) float v8f;

#define KTWO_PI 6.2831853071795864769f
#define KEPS    1e-6f
#define KLD     12   // padded inner dimension (rank-11 + 1 pad)

// ---------------------------------------------------------------------------
// Per-prediction-row precompute: F[n, 0..11] =
//   [A, 1, Ex0, Ex1, Ex2, S00, S11, S22, S01, S02, S12, 0]
// ---------------------------------------------------------------------------
__global__ void kent_pred_precompute(const float* __restrict__ kp,
                                     float* __restrict__ F, int N) {
  int n = blockIdx.x * blockDim.x + threadIdx.x;
  if (n >= N) return;
  float eta   = kp[n * 5 + 0];
  float alpha = kp[n * 5 + 1];
  float psi   = kp[n * 5 + 2];
  float kappa = kp[n * 5 + 3];
  float beta  = kp[n * 5 + 4];

  float sa, ca, se, ce, sp, cp;
  __sincosf(alpha, &sa, &ca);
  __sincosf(eta,   &se, &ce);
  __sincosf(psi,   &sp, &cp);

  // columns of Q (orthonormal frame)
  float g1x = ca,        g1y = sa * ce,                 g1z = sa * se;
  float g2x = -cp * sa,  g2y = cp * ca * ce - sp * se,  g2z = cp * ca * se + sp * ce;
  float g3x = sp * sa,   g3y = -sp * ca * ce - cp * se, g3z = -sp * ca * se + cp * ce;

  float km = kappa - 2.0f * beta;
  float kp2 = kappa + 2.0f * beta;
  float lkm = __logf(km), lkp = __logf(kp2);

  float c_a  = __logf(KTWO_PI) + kappa - 0.5f * __logf(km * kp2 + KEPS);
  float den15 = 1.5f * lkm + 1.5f * lkp + KEPS;
  float den25 = 2.5f * lkm + 2.5f * lkp + KEPS;
  float b2 = beta * beta;
  float c_ka = __logf(-KTWO_PI * (4.0f * b2 + kappa - kappa * kappa)) + kappa - den15;
  float k2 = kappa * kappa;
  float poly = k2 * k2 - 2.0f * k2 * kappa + (2.0f - 8.0f * b2) * k2
             + 8.0f * b2 * kappa + 16.0f * b2 * b2 + 4.0f * b2;
  float c_kk = __logf(KTWO_PI * poly) + kappa - den25;
  float c_cb = __logf(8.0f * (0.5f * KTWO_PI) * kappa) + __logf(beta) - den15;

  float l1  = __expf(c_ka - c_a);
  float ekk = __expf(c_kk - c_a);
  float eb  = __expf(c_cb - c_a);
  float l2  = 0.5f * (1.0f - ekk + eb);
  float l3  = 0.5f * (1.0f - ekk - eb);

  float Ex0 = l1 * g1x, Ex1 = l1 * g1y, Ex2 = l1 * g1z;

  // S = l1 g1 g1^T + l2 g2 g2^T + l3 g3 g3^T (symmetric)
  float S00 = l1 * g1x * g1x + l2 * g2x * g2x + l3 * g3x * g3x;
  float S11 = l1 * g1y * g1y + l2 * g2y * g2y + l3 * g3y * g3y;
  float S22 = l1 * g1z * g1z + l2 * g2z * g2z + l3 * g3z * g3z;
  float S01 = l1 * g1x * g1y + l2 * g2x * g2y + l3 * g3x * g3y;
  float S02 = l1 * g1x * g1z + l2 * g2x * g2z + l3 * g3x * g3z;
  float S12 = l1 * g1y * g1z + l2 * g2y * g2z + l3 * g3y * g3z;

  // q2 = g2^T S g2 ; q3 = g3^T S g3
  float q2 = S00 * g2x * g2x + S11 * g2y * g2y + S22 * g2z * g2z
           + 2.0f * (S01 * g2x * g2y + S02 * g2x * g2z + S12 * g2y * g2z);
  float q3 = S00 * g3x * g3x + S11 * g3y * g3y + S22 * g3z * g3z
           + 2.0f * (S01 * g3x * g3y + S02 * g3x * g3z + S12 * g3y * g3z);

  float dotEx = g1x * Ex0 + g1y * Ex1 + g1z * Ex2;
  float A = -c_a + kappa * dotEx + beta * (q2 - q3);

  float* f = F + (size_t)n * KLD;
  f[0] = A;    f[1] = 1.0f;
  f[2] = Ex0;  f[3] = Ex1;  f[4] = Ex2;
  f[5] = S00;  f[6] = S11;  f[7] = S22;
  f[8] = S01;  f[9] = S02;  f[10] = S12;
  f[11] = 0.0f;
}

// ---------------------------------------------------------------------------
// Per-target-row precompute: G[m, 0..11] =
//   [1, c_b, -kb*gb1(3), beta_b*(u3-u2)(6), 0]
// ---------------------------------------------------------------------------
__global__ void kent_targ_precompute(const float* __restrict__ kt,
                                     float* __restrict__ G, int M) {
  int m = blockIdx.x * blockDim.x + threadIdx.x;
  if (m >= M) return;
  float eta   = kt[m * 5 + 0];
  float alpha = kt[m * 5 + 1];
  float psi   = kt[m * 5 + 2];
  float kappa = kt[m * 5 + 3];
  float beta  = kt[m * 5 + 4];

  float sa, ca, se, ce, sp, cp;
  __sincosf(alpha, &sa, &ca);
  __sincosf(eta,   &se, &ce);
  __sincosf(psi,   &sp, &cp);

  float g1x = ca,        g1y = sa * ce,                 g1z = sa * se;
  float g2x = -cp * sa,  g2y = cp * ca * ce - sp * se,  g2z = cp * ca * se + sp * ce;
  float g3x = sp * sa,   g3y = -sp * ca * ce - cp * se, g3z = -sp * ca * se + cp * ce;

  float km = kappa - 2.0f * beta;
  float kp2 = kappa + 2.0f * beta;
  float c_b = __logf(KTWO_PI) + kappa - 0.5f * __logf(km * kp2 + KEPS);

  float* g = G + (size_t)m * KLD;
  g[0] = 1.0f;
  g[1] = c_b;
  g[2] = -kappa * g1x;
  g[3] = -kappa * g1y;
  g[4] = -kappa * g1z;
  g[5]  = beta * (g3x * g3x - g2x * g2x);
  g[6]  = beta * (g3y * g3y - g2y * g2y);
  g[7]  = beta * (g3z * g3z - g2z * g2z);
  g[8]  = 2.0f * beta * (g3x * g3y - g2x * g2y);
  g[9]  = 2.0f * beta * (g3x * g3z - g2x * g2z);
  g[10] = 2.0f * beta * (g3y * g3z - g2y * g2z);
  g[11] = 0.0f;
}

// ---------------------------------------------------------------------------
// Out = F (N x 12) * G^T (12 x M) via V_WMMA_F32_16X16X4_F32.
// One wave per 16x16 output tile; K=12 -> three K=4 WMMA steps, all f32.
// A-matrix lane layout (32-bit 16x4): lane L<16 holds (M=L, K=k0+0/1),
// lane L>=16 holds (M=L-16, K=k0+2/3) -> an 8B v2f load per lane per step.
// B (4x16) mirrors this with N=lane%16. C/D: VGPR v <-> row v+8*(lane/16).
// ---------------------------------------------------------------------------
__global__ void __launch_bounds__(32)
kent_wmma_gemm(const float* __restrict__ F, const float* __restrict__ G,
               float* __restrict__ out, int M) {
  const int lane = threadIdx.x;
  const int r = lane & 15;   // row within 16 / output column within tile
  const int h = lane >> 4;   // half-wave select
  const int m0 = blockIdx.x * 16;
  const int n0 = blockIdx.y * 16;

  const float* fr = F + (size_t)(n0 + r) * KLD + 2 * h;
  const float* gr = G + (size_t)(m0 + r) * KLD + 2 * h;

  v8f c = {};
#pragma unroll
  for (int k0 = 0; k0 < KLD; k0 += 4) {
    v2f a = *(const v2f*)(fr + k0);
    v2f b = *(const v2f*)(gr + k0);
    c = __builtin_amdgcn_wmma_f32_16x16x4_f32(
        /*neg_a=*/false, a, /*neg_b=*/false, b,
        /*c_mod=*/(short)0, c, /*reuse_a=*/false, /*reuse_b=*/false);
  }

  float* o = out + (size_t)(n0 + 8 * h) * M + m0 + r;
#pragma unroll
  for (int v = 0; v < 8; ++v) o[(size_t)v * M] = c[v];
}

// ---------------------------------------------------------------------------
extern "C" void kernel_launch(void* const* d_in, const int* in_sizes, int n_in,
                              void* d_out, int out_size, void* d_ws, size_t ws_size,
                              hipStream_t stream) {
  const float* kent_pred   = (const float*)d_in[0];
  const float* kent_target = (const float*)d_in[1];
  const int N = in_sizes[0] / 5;  // 2048
  const int M = in_sizes[1] / 5;  // 2048

  float* F = (float*)d_ws;                 // N x 12 floats
  float* G = F + (size_t)N * KLD;          // M x 12 floats

  kent_pred_precompute<<<(N + 255) / 256, 256, 0, stream>>>(kent_pred, F, N);
  kent_targ_precompute<<<(M + 255) / 256, 256, 0, stream>>>(kent_target, G, M);

  dim3 grid(M / 16, N / 16);               // 128 x 128 tiles, one wave each
  kent_wmma_gemm<<<grid, 32, 0, stream>>>(F, G, (float*)d_out, M);
}